// StatDenoiser_54640573939938
// MI455X (gfx1250) — compile-verified
//
#include <hip/hip_runtime.h>
#include <math.h>

typedef float v2f __attribute__((ext_vector_type(2)));
typedef float v8f __attribute__((ext_vector_type(8)));

#define RAD  5
#define KSZ  11
#define TILE 16
#define TS   26   // TILE + 2*RAD

__global__ __launch_bounds__(256) void StatDenoiser_kernel(
    const float* __restrict__ img,   // (3,H,W)
    const float* __restrict__ gdn,   // (8,H,W)
    const float* __restrict__ est,   // (3,H,W)
    const float* __restrict__ var,   // (3,H,W)
    const int*   __restrict__ sppp,  // scalar
    float*       __restrict__ out,   // (3,H,W)
    int H, int W)
{
  struct Smem {
    float h[8][TS][TS];          // sqrt(sigma_inv)*guidance, halo tile
    float q[TS][TS];             // sum_c h^2
    float e[3][TS][TS];          // estimands
    float v[3][TS][TS];          // variances
    float im[3][TS][TS];         // image
    float rbuf[KSZ][TILE][12];   // banded cross-corr; slot 11 = write dump
    float part[TILE][KSZ][4];    // per-(pixel,dy) partial {wsum, acc0..2}
  };
  __shared__ Smem sm;

  const int tid = threadIdx.x;
  const int X0 = blockIdx.x * TILE;
  const int Y0 = blockIdx.y * TILE;
  const int HW = H * W;

  // sqrt of SIGMA_INV = [0.1,0.1,50,50,50,10,10,10]
  const float rs[8] = {0.316227766017f, 0.316227766017f,
                       7.071067811865f, 7.071067811865f, 7.071067811865f,
                       3.162277660168f, 3.162277660168f, 3.162277660168f};

  // gamma_w = Cornish-Fisher t-quantile at p = 1 - ALPHA/2 = 0.9975, df = 2*spp-2
  const int   spp = sppp[0];
  const float df  = 2.0f * (float)spp - 2.0f;
  const float z   = 2.8070337683f;            // ndtri(0.9975)
  const float z2 = z*z, z3 = z2*z, z5 = z3*z2, z7 = z5*z2;
  const float g1 = (z3 + z) / (4.0f * df);
  const float g2 = (5.0f*z5 + 16.0f*z3 + 3.0f*z) / (96.0f * df * df);
  const float g3 = (3.0f*z7 + 19.0f*z5 + 17.0f*z3 - 15.0f*z) / (384.0f * df*df*df);
  const float gam = z + g1 + g2 + g3;
  // t_stat < gamma  <=>  w < wthr   (w in [0.5,1]; w==1 -> fail since wthr < 1)
  const float wthr = 1.0f - 0.5f / (gam*gam + 1.0f);

  // ---------------- Phase 1: load halo tile (zero-padded) ----------------
  for (int i = tid; i < TS*TS; i += 256) {
    int ly = i / TS, lx = i - ly*TS;
    int gy = Y0 + ly - RAD, gx = X0 + lx - RAD;
    bool inb = (gy >= 0) & (gy < H) & (gx >= 0) & (gx < W);
    int base = gy * W + gx;
    float qa = 0.0f;
    #pragma unroll
    for (int c = 0; c < 8; ++c) {
      float gv = inb ? gdn[c*HW + base] : 0.0f;
      float hv = gv * rs[c];
      sm.h[c][ly][lx] = hv;
      qa += hv * hv;
    }
    sm.q[ly][lx] = qa;
    #pragma unroll
    for (int c = 0; c < 3; ++c) {
      sm.e[c][ly][lx]  = inb ? est[c*HW + base] : 0.0f;
      sm.v[c][ly][lx]  = inb ? var[c*HW + base] : 0.0f;
      sm.im[c][ly][lx] = inb ? img[c*HW + base] : 0.0f;
    }
  }
  __syncthreads();

  const int lane = tid & 31;
  const int wv   = tid >> 5;     // wave id (wave32, 8 waves)
  const int lm   = lane & 15;    // M (A) / N (B,C) index
  const int lh   = lane >> 4;    // half of wave

  const int xx  = tid & 15;      // pixel x within tile
  const int dyi = tid >> 4;      // dy task, active if < 11

  for (int yy = 0; yy < TILE; ++yy) {
    // ------------- Phase 2: WMMA cross-correlation r(p, dy, dx) -------------
    // A[m][k] = h[k][yy+RAD][RAD+m]  (16x8, two K=4 chunks)
    // 32-bit A 16x4 layout: VGPR j, half lh -> K = j + 2*lh
    v2f a0, a1;
    {
      int ch = 2*lh;
      a0.x = sm.h[ch+0][yy+RAD][RAD+lm];
      a0.y = sm.h[ch+1][yy+RAD][RAD+lm];
      a1.x = sm.h[ch+4][yy+RAD][RAD+lm];
      a1.y = sm.h[ch+5][yy+RAD][RAD+lm];
    }
    for (int task = wv; task < 22; task += 8) {   // uniform per wave: EXEC all 1s at WMMA
      int dy  = task % 11;
      int nt  = task / 11;
      int col = nt*16 + lm;                        // neighbor x in tile frame, 0..31
      bool cin = (col < TS);
      int colc = cin ? col : (TS - 1);             // clamp: straight-line loads + cndmask
      int kb  = 2*lh;
      float t0 = sm.h[kb+0][yy+dy][colc];
      float t1 = sm.h[kb+1][yy+dy][colc];
      float t2 = sm.h[kb+4][yy+dy][colc];
      float t3 = sm.h[kb+5][yy+dy][colc];
      v2f b0, b1;
      b0.x = cin ? t0 : 0.0f;
      b0.y = cin ? t1 : 0.0f;
      b1.x = cin ? t2 : 0.0f;
      b1.y = cin ? t3 : 0.0f;
      v8f cc = {0.0f,0.0f,0.0f,0.0f,0.0f,0.0f,0.0f,0.0f};
      cc = __builtin_amdgcn_wmma_f32_16x16x4_f32(false, a0, false, b0, (short)0, cc, false, false);
      cc = __builtin_amdgcn_wmma_f32_16x16x4_f32(false, a1, false, b1, (short)0, cc, false, false);
      // Banded scatter: C[m][n], lane n=lm(+16*nt), m = j + 8*lh.
      // Out-of-band lanes dump into slot 11 (never read) -> no EXEC churn.
      #pragma unroll
      for (int j = 0; j < 8; ++j) {
        int m = j + 8*lh;
        int d = col - m;                           // band offset dx
        int dd = ((unsigned)d < (unsigned)KSZ && cin) ? d : KSZ;
        sm.rbuf[dy][m][dd] = cc[j];
      }
    }
    __syncthreads();

    // ------------- Phase 3: per-(pixel, dy) statistics -------------
    if (dyi < KSZ) {
      float qc  = sm.q[yy+RAD][xx+RAD];
      float ce[3], cv[3];
      #pragma unroll
      for (int c = 0; c < 3; ++c) {
        ce[c] = sm.e[c][yy+RAD][xx+RAD];
        cv[c] = sm.v[c][yy+RAD][xx+RAD];
      }
      float wsum = 0.0f, A0 = 0.0f, A1 = 0.0f, A2 = 0.0f;
      for (int dx = 0; dx < KSZ; ++dx) {
        int ny = yy + dyi, nx = xx + dx;
        float r  = sm.rbuf[dyi][xx][dx];
        float S  = qc + sm.q[ny][nx] - 2.0f * r;   // Mahalanobis guidance distance
        float bw = __expf(-0.5f * S);
        bool pass = true;
        #pragma unroll
        for (int c = 0; c < 3; ++c) {
          float se = sm.e[c][ny][nx], sv = sm.v[c][ny][nx];
          float d  = ce[c] - se;
          float d2 = d * d;
          float num = 2.0f*d2 + cv[c] + sv;
          float den = num + cv[c] + sv;            // = 2*(d2+cv+sv)
          float w = (den == 0.0f) ? 0.5f : num / den;
          if (((cv[c] == 0.0f) || (sv == 0.0f)) && (ce[c] != se)) w = 1.0f;
          pass = pass && (w < wthr);
        }
        if ((dyi == RAD) && (dx == RAD)) pass = true;  // center always member
        float fw = pass ? bw : 0.0f;
        wsum += fw;
        A0 += fw * sm.im[0][ny][nx];
        A1 += fw * sm.im[1][ny][nx];
        A2 += fw * sm.im[2][ny][nx];
      }
      sm.part[xx][dyi][0] = wsum;
      sm.part[xx][dyi][1] = A0;
      sm.part[xx][dyi][2] = A1;
      sm.part[xx][dyi][3] = A2;
    }
    __syncthreads();

    // ------------- Phase 4: reduce over dy, normalize, store -------------
    if (tid < TILE) {
      float ws = 0.0f, B0 = 0.0f, B1 = 0.0f, B2 = 0.0f;
      #pragma unroll
      for (int d = 0; d < KSZ; ++d) {
        ws += sm.part[tid][d][0];
        B0 += sm.part[tid][d][1];
        B1 += sm.part[tid][d][2];
        B2 += sm.part[tid][d][3];
      }
      float inv = 1.0f / fmaxf(ws, 1e-10f);
      int gy = Y0 + yy, gx = X0 + tid;
      if (gy < H && gx < W) {
        out[0*HW + gy*W + gx] = B0 * inv;
        out[1*HW + gy*W + gx] = B1 * inv;
        out[2*HW + gy*W + gx] = B2 * inv;
      }
    }
    __syncthreads();   // rbuf/part reused next yy iteration
  }
}

extern "C" void kernel_launch(void* const* d_in, const int* in_sizes, int n_in,
                              void* d_out, int out_size, void* d_ws, size_t ws_size,
                              hipStream_t stream) {
  (void)n_in; (void)out_size; (void)d_ws; (void)ws_size;
  const float* img = (const float*)d_in[0];
  const float* gdn = (const float*)d_in[1];
  const float* est = (const float*)d_in[2];
  const float* var = (const float*)d_in[3];
  const int*   spp = (const int*)d_in[4];

  int hw = in_sizes[0] / 3;                 // image is (1,3,H,W), square
  int W = (int)(sqrt((double)hw) + 0.5);
  int H = (W > 0) ? (hw / W) : 0;

  dim3 grid((W + TILE - 1) / TILE, (H + TILE - 1) / TILE);
  StatDenoiser_kernel<<<grid, 256, 0, stream>>>(img, gdn, est, var, spp,
                                                (float*)d_out, H, W);
}